// SymmetricContraction_63634235457558
// MI455X (gfx1250) — compile-verified
//
#include <hip/hip_runtime.h>

typedef float f32x2 __attribute__((ext_vector_type(2)));
typedef float f32x4 __attribute__((ext_vector_type(4)));
typedef float f32x8 __attribute__((ext_vector_type(8)));

// Input layout (setup_inputs dict order, lists flattened depth-first):
//  0: A           (1024,128,16) f32
//  1: node_types  (1024,10)     f32 one-hot
//  2+2l: Us[l][0] ((2l+1)?,16,2)      nu=1   (l=0: (16,2))
//  3+2l: Us[l][1] ((2l+1)?,16,16,6)   nu=2   (l=0: (16,16,6))
// 10+2l: Ws[l][0] (10,2,128)
// 11+2l: Ws[l][1] (10,6,128)
// out: (1024,128,16) f32

struct SymParams {
    const float* __restrict__ A;
    const float* __restrict__ NT;
    const float* __restrict__ U1[4];
    const float* __restrict__ U2[4];
    const float* __restrict__ W1[4];
    const float* __restrict__ W2[4];
    float* __restrict__ out;
};

__global__ __launch_bounds__(256)
void SymmetricContraction_wmma_f32(SymParams p)
{
    const int b    = blockIdx.x;
    const int tid  = threadIdx.x;
    const int wave = tid >> 5;        // 8 waves -> 8 channel tiles of 16
    const int lane = tid & 31;
    const int j    = lane & 15;       // matrix row (A-op) / column (B-op)
    const bool hi  = lane >= 16;      // K-half selector per WMMA layout
    const int c    = (wave << 4) | j; // channel 0..127

    // one-hot node type -> index e
    int e = 0;
    #pragma unroll
    for (int q = 0; q < 10; ++q)
        if (p.NT[b * 10 + q] > 0.5f) e = q;

    // a[i] = A[b,c,i]
    const float* Arow = p.A + (size_t)(b * 128 + c) * 16;
    float a[16];
    #pragma unroll
    for (int i = 0; i < 16; i += 4) {
        f32x4 v = *(const f32x4*)(Arow + i);
        a[i] = v.x; a[i+1] = v.y; a[i+2] = v.z; a[i+3] = v.w;
    }
    // per-lane a[j']-values matching D-tile row ownership (rows v / v+8)
    float areg[8];
    #pragma unroll
    for (int v = 0; v < 8; ++v) areg[v] = hi ? a[v + 8] : a[v];

    // per-degree weights for (e, c)
    float w2l[4][6], w1l[4][2];
    #pragma unroll
    for (int l = 0; l < 4; ++l) {
        #pragma unroll
        for (int k = 0; k < 6; ++k) w2l[l][k] = p.W2[l][(e * 6 + k) * 128 + c];
        #pragma unroll
        for (int k = 0; k < 2; ++k) w1l[l][k] = p.W1[l][(e * 2 + k) * 128 + c];
    }

    // output-row -> (degree l, local m) map
    static constexpr int LOF[16]  = {0, 1,1,1, 2,2,2,2,2, 3,3,3,3,3,3,3};
    static constexpr int MLOC[16] = {0, 0,1,2, 0,1,2,3,4, 0,1,2,3,4,5,6};

    float* outrow = p.out + (size_t)(b * 128 + c) * 16;
    const int hi2 = hi ? 2 : 0;

    #pragma unroll
    for (int t = 0; t < 16; ++t) {
        const int l = LOF[t];
        const int m = MLOC[t];
        // U2_l[m, j, i, k] flat = m*1536 + j*96 + i*6 + k ; q = i*6+k
        const float* u2row = p.U2[l] + m * 1536 + j * 96 + hi2;
        const float* u1row = p.U1[l] + m * 32 + j * 2;

        f32x8 D = {0.f, 0.f, 0.f, 0.f, 0.f, 0.f, 0.f, 0.f};

        // 24 K-steps over q = 0..95  (q = i*6+k), K=4 per WMMA
        #pragma unroll
        for (int s = 0; s < 24; ++s) {
            // A-operand: lanes 0-15 take K=4s,4s+1; lanes 16-31 take 4s+2,4s+3
            f32x2 aop = *(const f32x2*)(u2row + 4 * s);
            const int q = 4 * s;
            const float x0 = a[(q + 0) / 6] * w2l[l][(q + 0) % 6];
            const float x1 = a[(q + 1) / 6] * w2l[l][(q + 1) % 6];
            const float x2 = a[(q + 2) / 6] * w2l[l][(q + 2) % 6];
            const float x3 = a[(q + 3) / 6] * w2l[l][(q + 3) % 6];
            f32x2 bop;
            bop.x = hi ? x2 : x0;
            bop.y = hi ? x3 : x1;
            D = __builtin_amdgcn_wmma_f32_16x16x4_f32(
                    false, aop, false, bop, (short)0, D, false, false);
        }

        // nu=1 tail: K columns 96,97 (valid only in low K-half; high half = 0)
        {
            f32x2 lv = *(const f32x2*)(u1row);
            f32x2 aop, bop;
            aop.x = hi ? 0.f : lv.x;
            aop.y = hi ? 0.f : lv.y;
            bop.x = hi ? 0.f : w1l[l][0];
            bop.y = hi ? 0.f : w1l[l][1];
            D = __builtin_amdgcn_wmma_f32_16x16x4_f32(
                    false, aop, false, bop, (short)0, D, false, false);
        }

        // out[b,c,t] = sum_j c_tensor[t,j] * a[j]
        // lane (col c) owns rows v (low half) / v+8 (high half) in D[v]
        float part = 0.f;
        #pragma unroll
        for (int v = 0; v < 8; ++v) part += D[v] * areg[v];
        part += __shfl_xor(part, 16, 32);   // combine j=0..7 with j=8..15
        if (!hi) outrow[t] = part;
    }
}

extern "C" void kernel_launch(void* const* d_in, const int* in_sizes, int n_in,
                              void* d_out, int out_size, void* d_ws, size_t ws_size,
                              hipStream_t stream)
{
    SymParams p;
    p.A  = (const float*)d_in[0];
    p.NT = (const float*)d_in[1];
    for (int l = 0; l < 4; ++l) {
        p.U1[l] = (const float*)d_in[2 + 2 * l];
        p.U2[l] = (const float*)d_in[3 + 2 * l];
        p.W1[l] = (const float*)d_in[10 + 2 * l];
        p.W2[l] = (const float*)d_in[11 + 2 * l];
    }
    p.out = (float*)d_out;

    const int n_nodes = in_sizes[0] / (128 * 16);   // 1024
    SymmetricContraction_wmma_f32<<<n_nodes, 256, 0, stream>>>(p);
}